// NearestNeighborTokenizer_9002251452832
// MI455X (gfx1250) — compile-verified
//
#include <hip/hip_runtime.h>

// CDNA5 / gfx1250 fused NN-tokenizer:
//   d2[b,n,m] = ||x||^2 + ||c||^2 - 2 x.c ; out = (min_m d2 <= 0.1) ? argmin_m : -1
// Exact-fp32 GEMM on the matrix pipe (V_WMMA_F32_16X16X4_F32), fused running
// min/argmin in registers (268M-element d2 never materialized), codes staged
// through padded LDS shared by 8 waves/block, per-stage cooperative ||c||^2.

typedef __attribute__((ext_vector_type(2))) float v2f;
typedef __attribute__((ext_vector_type(4))) float v4f;
typedef __attribute__((ext_vector_type(8))) float v8f;

#define WAVES_PER_BLOCK 8
#define STAGE_CODES     128
#define ROW_STRIDE      68      // floats; 64 + 4 pad -> conflict-free ds_load_b64
#define DIST_THRESHOLD  0.1f

__global__ __launch_bounds__(256)
void nn_tokenizer_wmma_f32(const float* __restrict__ x,      // [65536, 64]
                           const float* __restrict__ codes,  // [n_codes, 64]
                           float* __restrict__ out,          // [65536]
                           int n_codes)
{
    __shared__ float ldsB[STAGE_CODES * ROW_STRIDE];          // ~34 KB
    __shared__ float ldsC2[STAGE_CODES];                      // per-stage ||c||^2
    __shared__ float x2buf[WAVES_PER_BLOCK][16];

    const int lane = threadIdx.x & 31;
    const int wave = threadIdx.x >> 5;
    const int half = lane >> 4;        // 0: K pair {0,1}, 1: K pair {2,3}
    const int l16  = lane & 15;

    const int row_block = blockIdx.x * WAVES_PER_BLOCK + wave;
    const int m_base    = row_block * 16;   // first global x-row of this wave

    // ---- Load A tile (16 rows x 64 dims) into WMMA 16x4 f32 layout -------
    // lane%16 = M, lane-half selects K offset {0,1} vs {2,3} within each K4 chunk.
    v2f a[16];
    const float* xrow = x + (size_t)(m_base + l16) * 64 + 2 * half;
    float x2part = 0.f;
#pragma unroll
    for (int kk = 0; kk < 16; ++kk) {
        a[kk] = *(const v2f*)(xrow + 4 * kk);
        x2part += a[kk].x * a[kk].x + a[kk].y * a[kk].y;
    }
    float x2full = x2part + __shfl_xor(x2part, 16, 32);  // combine K halves
    if (half == 0) x2buf[wave][l16] = x2full;            // same-wave consumer

    // Running min of e = ||c||^2 - 2*dot (= d2 - ||x||^2, same argmin)
    float beste[8];
    int   besti[8];
#pragma unroll
    for (int v = 0; v < 8; ++v) { beste[v] = 3.4e38f; besti[v] = 0; }

    const int nstages = n_codes / STAGE_CODES;
    for (int s = 0; s < nstages; ++s) {
        __syncthreads();   // previous stage's readers done before overwrite
        // ---- cooperative stage: 128 codes -> padded LDS (float4 moves) ----
        const float* src = codes + (size_t)s * STAGE_CODES * 64;
#pragma unroll
        for (int i = 0; i < 8; ++i) {
            int g    = threadIdx.x + i * 256;   // float4 index, [0, 2048)
            int row  = g >> 4;
            int col4 = g & 15;
            v4f val  = *(const v4f*)(src + (size_t)g * 4);
            *(v4f*)(&ldsB[row * ROW_STRIDE + col4 * 4]) = val;
        }
        // warm next stage's 32 KB tile (global_prefetch_b8; 128B per thread)
        if (s + 1 < nstages)
            __builtin_prefetch(codes + (size_t)(s + 1) * STAGE_CODES * 64 +
                               (size_t)threadIdx.x * 32, 0, 0);
        __syncthreads();

        // ---- per-stage ||c||^2: first 128 threads, one code row each ------
        if (threadIdx.x < STAGE_CODES) {
            const float* r = &ldsB[threadIdx.x * ROW_STRIDE];
            float acc = 0.f;
#pragma unroll
            for (int q = 0; q < 16; ++q) {
                v4f t = *(const v4f*)(r + 4 * q);
                acc += t.x * t.x + t.y * t.y + t.z * t.z + t.w * t.w;
            }
            ldsC2[threadIdx.x] = acc;
        }
        __syncthreads();

#pragma unroll 1
        for (int chunk = 0; chunk < STAGE_CODES / 16; ++chunk) {
            const int n_local = chunk * 16 + l16;
            const float c2 = ldsC2[n_local];            // 1 ds_load_b32, broadcast
            const float* brow = &ldsB[n_local * ROW_STRIDE + 2 * half];
            v2f b[16];
#pragma unroll
            for (int kk = 0; kk < 16; ++kk)
                b[kk] = *(const v2f*)(brow + 4 * kk);

            // ---- 16 chained fp32 WMMAs: dot = x_tile . codes_tile^T ------
            v8f c = {};
#pragma unroll
            for (int kk = 0; kk < 16; ++kk)
                c = __builtin_amdgcn_wmma_f32_16x16x4_f32(
                        false, a[kk], false, b[kk],
                        (short)0, c, false, false);

            const int n_glob = s * STAGE_CODES + chunk * 16 + l16;
#pragma unroll
            for (int v = 0; v < 8; ++v) {
                float e = c2 - 2.0f * c[v];
                if (e < beste[v]) { beste[v] = e; besti[v] = n_glob; } // first-min wins
            }
        }
    }

    // ---- reduce (min,argmin) across the 16 lanes of each half ------------
#pragma unroll
    for (int v = 0; v < 8; ++v) {
        float e = beste[v];
        int   idx = besti[v];
#pragma unroll
        for (int m = 8; m >= 1; m >>= 1) {
            float eo = __shfl_xor(e, m, 32);
            int   io = __shfl_xor(idx, m, 32);
            if (eo < e || (eo == e && io < idx)) { e = eo; idx = io; }
        }
        beste[v] = e; besti[v] = idx;
    }

    // lane 0 holds rows m=0..7 (vgpr v -> M=v); lane 16 holds m=8..15 (M=v+8)
    if (l16 == 0) {
#pragma unroll
        for (int v = 0; v < 8; ++v) {
            int   m  = v + 8 * half;
            float d2 = x2buf[wave][m] + beste[v];
            out[m_base + m] = (d2 <= DIST_THRESHOLD) ? (float)besti[v] : -1.0f;
        }
    }
}

extern "C" void kernel_launch(void* const* d_in, const int* in_sizes, int n_in,
                              void* d_out, int out_size, void* d_ws, size_t ws_size,
                              hipStream_t stream) {
    const float* x     = (const float*)d_in[0];   // [16,4096,64] fp32
    const float* codes = (const float*)d_in[1];   // [4096,64]    fp32
    float* out = (float*)d_out;                   // [16*4096]

    const int n_codes    = in_sizes[1] / 64;      // 4096
    const int total_rows = in_sizes[0] / 64;      // 65536
    const int blocks     = total_rows / (16 * WAVES_PER_BLOCK); // 512

    nn_tokenizer_wmma_f32<<<blocks, 256, 0, stream>>>(x, codes, out, n_codes);
}